// SoftDTW_12103217840508
// MI455X (gfx1250) — compile-verified
//
#include <hip/hip_runtime.h>
#include <hip/hip_bf16.h>

typedef __attribute__((ext_vector_type(2))) float v2f;
typedef __attribute__((ext_vector_type(8))) float v8f;

#define SDTW_B 256
#define SDTW_N 256
#define SDTW_M 256
#define SDTW_D 32
#define SDTW_BIG 1e8f

// ---------------------------------------------------------------------------
// Kernel 1: batched squared-distance matrix via f32 WMMA.
// dist[b,i,j] = max(0, |x_i|^2 + |y_j|^2 - 2 * x_i . y_j)
// One wave computes one 16x16 output tile; K=32 -> 8 x V_WMMA_F32_16X16X4_F32.
// ---------------------------------------------------------------------------
__global__ __launch_bounds__(256) void sdtw_dist_wmma(
    const float* __restrict__ x,   // [B, N, D]
    const float* __restrict__ y,   // [B, M, D]
    float* __restrict__ dist)      // [B, N, M]
{
    const int wave = threadIdx.x >> 5;               // 0..7
    const int lane = threadIdx.x & 31;
    const int tile = blockIdx.x * 8 + wave;          // global 16x16 tile id
    const int b  = tile >> 8;                        // 256 tiles per batch
    const int t  = tile & 255;
    const int i0 = (t >> 4) * 16;
    const int j0 = (t & 15) * 16;

    const float* xb = x + ((size_t)b * SDTW_N) * SDTW_D;
    const float* yb = y + ((size_t)b * SDTW_M) * SDTW_D;

    const int half = lane >> 4;     // 0 or 1
    const int lrow = lane & 15;

    v8f acc = {};
#pragma unroll
    for (int k0 = 0; k0 < SDTW_D; k0 += 4) {
        // A frag: lane holds x[i0+lrow][k0 + 2*half + {0,1}]
        const float* ap = xb + (size_t)(i0 + lrow) * SDTW_D + k0 + 2 * half;
        v2f a; a.x = ap[0]; a.y = ap[1];
        // B frag (B = Y^T, so B[k][n] = y[n][k]): same indexing into y
        const float* bp = yb + (size_t)(j0 + lrow) * SDTW_D + k0 + 2 * half;
        v2f bf; bf.x = bp[0]; bf.y = bp[1];
        acc = __builtin_amdgcn_wmma_f32_16x16x4_f32(
            /*neg_a=*/false, a, /*neg_b=*/false, bf,
            /*c_mod=*/(short)0, acc, /*reuse_a=*/false, /*reuse_b=*/false);
    }

    // Squared norms: lanes 0..15 compute |x_{i0+lane}|^2, lanes 16..31 |y_{j0+lane-16}|^2
    float s = 0.0f;
    const float* p = (half == 0) ? (xb + (size_t)(i0 + lrow) * SDTW_D)
                                 : (yb + (size_t)(j0 + lrow) * SDTW_D);
#pragma unroll
    for (int k = 0; k < SDTW_D; ++k) { float v = p[k]; s = fmaf(v, v, s); }

    // C/D layout: lane L, vgpr v -> (M = v + 8*(L>>4), N = L&15)
#pragma unroll
    for (int v = 0; v < 8; ++v) {
        const int M = v + 8 * half;
        const int N = lrow;
        const float x2 = __shfl(s, M, 32);        // from lane M   (< 16)
        const float y2 = __shfl(s, 16 + N, 32);   // from lane 16+N
        float dd = x2 + y2 - 2.0f * acc[v];
        dd = fmaxf(dd, 0.0f);
        dist[((size_t)b * SDTW_N + (i0 + M)) * SDTW_M + (j0 + N)] = dd;
    }
}

// ---------------------------------------------------------------------------
// Kernel 2: soft-DTW DP, one workgroup per batch, anti-diagonal wavefront.
// Thread t owns row i = t+1. Three rotating diagonal buffers in LDS.
// D[i][j] = dist[i-1][j-1] + softmin(D[i-1][j], D[i][j-1], D[i-1][j-1])
// ---------------------------------------------------------------------------
__global__ __launch_bounds__(256) void sdtw_dp(
    const float* __restrict__ dist,   // [B, N, M]
    float* __restrict__ partial)      // [B]
{
    const int b = blockIdx.x;
    const int t = threadIdx.x;        // 0..255
    const int i = t + 1;              // row 1..256

    __shared__ float buf[3][SDTW_N + 1];

    // diag 0: D[0][0] = 0; diag 1: D[0][1] = D[1][0] = BIG
    buf[0][t] = SDTW_BIG;
    buf[1][t] = SDTW_BIG;
    buf[2][t] = SDTW_BIG;
    if (t == 0) {
        buf[0][SDTW_N] = SDTW_BIG;
        buf[1][SDTW_N] = SDTW_BIG;
        buf[2][SDTW_N] = SDTW_BIG;
        buf[0][0] = 0.0f;
    }
    __syncthreads();

    const float* db = dist + (size_t)b * SDTW_N * SDTW_M;
    float result = 0.0f;

    for (int k = 2; k <= SDTW_N + SDTW_M; ++k) {
        float* prev2 = buf[k % 3];          // diag k-2 lives where cur will go? no:
        // rotate: (k-2)%3, (k-1)%3, k%3
        prev2 = buf[(k - 2) % 3];
        float* prev1 = buf[(k - 1) % 3];
        float* cur   = buf[k % 3];

        const int j = k - i;
        if (j >= 1 && j <= SDTW_M) {
            const float dij = db[(size_t)(i - 1) * SDTW_M + (j - 1)];
            const float a  = prev1[i - 1];  // D[i-1][j]
            const float bb = prev1[i];      // D[i][j-1]
            const float c  = prev2[i - 1];  // D[i-1][j-1]
            const float mn  = fminf(a, fminf(bb, c));
            const float sum = __expf(mn - a) + __expf(mn - bb) + __expf(mn - c);
            const float val = dij + (mn - __logf(sum));   // gamma = 1
            cur[i] = val;
            if (i == SDTW_N && j == SDTW_M) result = val;
        }
        if (t == 0) cur[0] = SDTW_BIG;                    // D[0][k] boundary
        if (i == k && k <= SDTW_N) cur[i] = SDTW_BIG;     // D[k][0] boundary
        __syncthreads();
    }

    if (t == SDTW_N - 1) partial[b] = result;   // captured at (n, m)
}

// ---------------------------------------------------------------------------
// Kernel 3: mean over B partials -> scalar output.
// ---------------------------------------------------------------------------
__global__ __launch_bounds__(256) void sdtw_reduce(
    const float* __restrict__ partial, float* __restrict__ out)
{
    __shared__ float s[SDTW_B];
    const int t = threadIdx.x;
    s[t] = partial[t];
    __syncthreads();
    for (int off = SDTW_B / 2; off > 0; off >>= 1) {
        if (t < off) s[t] += s[t + off];
        __syncthreads();
    }
    if (t == 0) out[0] = s[0] / (float)SDTW_B;
}

extern "C" void kernel_launch(void* const* d_in, const int* in_sizes, int n_in,
                              void* d_out, int out_size, void* d_ws, size_t ws_size,
                              hipStream_t stream) {
    (void)in_sizes; (void)n_in; (void)out_size; (void)ws_size;
    const float* x = (const float*)d_in[0];   // [256,256,32] f32
    const float* y = (const float*)d_in[1];   // [256,256,32] f32
    float* dist    = (float*)d_ws;                                  // 64 MB
    float* partial = dist + (size_t)SDTW_B * SDTW_N * SDTW_M;       // +1 KB
    float* out     = (float*)d_out;

    // 65536 16x16 tiles, 8 waves (tiles) per 256-thread block
    sdtw_dist_wmma<<<(SDTW_B * 256) / 8, 256, 0, stream>>>(x, y, dist);
    sdtw_dp<<<SDTW_B, 256, 0, stream>>>(dist, partial);
    sdtw_reduce<<<1, SDTW_B, 0, stream>>>(partial, out);
}